// SGraphAttentionLayer_23965917512151
// MI455X (gfx1250) — compile-verified
//
#include <hip/hip_runtime.h>
#include <hip/hip_bf16.h>

#ifndef __has_builtin
#define __has_builtin(x) 0
#endif

typedef __attribute__((ext_vector_type(2))) float v2f;
typedef __attribute__((ext_vector_type(8))) float v8f;

#if __has_builtin(__builtin_amdgcn_wmma_f32_16x16x4_f32)
#define SGAT_HAVE_WMMA_F32 1
#else
#define SGAT_HAVE_WMMA_F32 0
#endif

// ---------------------------------------------------------------------------
// Kernel 1: zero the per-node accumulators (sums[N*64], s_ea[N], cnt[N]).
// ---------------------------------------------------------------------------
__global__ void sgat_zero_kernel(float* __restrict__ p, long long n) {
    long long i = (long long)blockIdx.x * blockDim.x + threadIdx.x;
    if (i < n) p[i] = 0.0f;
}

// ---------------------------------------------------------------------------
// Kernel 2: Y[N,128] = X[N,64] @ Wt[64,128]  (Wt = [W_top | W_bot])
// One wave per 16(node) x 16(outch) tile using V_WMMA_F32_16X16X4_F32.
//   A (16x4 f32): lane l<16 holds row M=l, {K0,K0+1}; lane l+16 holds {K0+2,K0+3}
//   B (4x16 f32): symmetric with N = lane&15
//   C/D (16x16 f32): VGPR r holds M=r (lanes 0-15) / M=r+8 (lanes 16-31), N=lane&15
// ---------------------------------------------------------------------------
__global__ void sgat_gemm_kernel(const float* __restrict__ x,
                                 const float* __restrict__ w,
                                 float* __restrict__ Y, int N) {
    const int lane  = threadIdx.x;          // 0..31
    const int wv    = threadIdx.y;          // 0..3
    const int ntile = blockIdx.x;           // node tile
    const int otile = blockIdx.y * 4 + wv;  // out-channel tile 0..7
    const int mbase = ntile * 16;
    const int cbase = otile * 16;           // out channel base 0..112
    const int l15   = lane & 15;
    const int lh    = lane >> 4;            // 0 or 1

    // B column pointer: Wt[k][oc] = oc<64 ? w[k*64+oc] : w[(64+k)*64 + (oc-64)]
    const int oc = cbase + l15;
    const float* wb = (oc < 64) ? (w + oc) : (w + 64 * 64 + (oc - 64));

    // A row pointer (clamped so EXEC stays uniform for WMMA)
    int arow = mbase + l15;
    if (arow >= N) arow = N - 1;
    const float* xa = x + (long long)arow * 64;

#if SGAT_HAVE_WMMA_F32
    v8f acc = {};
#pragma unroll
    for (int kk = 0; kk < 16; ++kk) {
        const int k0 = kk * 4 + lh * 2;
        v2f a; a.x = xa[k0];          a.y = xa[k0 + 1];
        v2f b; b.x = wb[k0 * 64];     b.y = wb[(k0 + 1) * 64];
        acc = __builtin_amdgcn_wmma_f32_16x16x4_f32(
            false, a, false, b, (short)0, acc, false, false);
    }
#pragma unroll
    for (int r = 0; r < 8; ++r) {
        const int m = r + 8 * lh;
        const int node = mbase + m;
        if (node < N) Y[(long long)node * 128 + cbase + l15] = acc[r];
    }
#else
    // Scalar fallback with identical output mapping (no WMMA builtin found).
    float acc[8];
#pragma unroll
    for (int r = 0; r < 8; ++r) {
        const int m = r + 8 * lh;
        int node = mbase + m;
        if (node >= N) node = N - 1;
        const float* xr = x + (long long)node * 64;
        float s = 0.0f;
        for (int k = 0; k < 64; ++k) s = fmaf(xr[k], wb[k * 64], s);
        acc[r] = s;
    }
#pragma unroll
    for (int r = 0; r < 8; ++r) {
        const int m = r + 8 * lh;
        const int node = mbase + m;
        if (node < N) Y[(long long)node * 128 + cbase + l15] = acc[r];
    }
#endif
}

// ---------------------------------------------------------------------------
// Kernel 3: edge scatter. 64 threads per edge (one per channel).
//   sums[row] += ea * y2[col];  s_ea[row] += ea;  cnt[row] += 1
// Native f32 atomics via __hip_atomic_fetch_add (relaxed, agent scope).
// ---------------------------------------------------------------------------
__device__ __forceinline__ void sgat_atomic_add(float* p, float v) {
    (void)__hip_atomic_fetch_add(p, v, __ATOMIC_RELAXED, __HIP_MEMORY_SCOPE_AGENT);
}

__global__ void sgat_edge_kernel(const long long* __restrict__ ei,
                                 const float* __restrict__ ea,
                                 const float* __restrict__ Y,
                                 float* __restrict__ sums,
                                 float* __restrict__ s_ea,
                                 float* __restrict__ cnt,
                                 long long E) {
    long long idx = (long long)blockIdx.x * blockDim.x + threadIdx.x;
    long long e = idx >> 6;
    if (e >= E) return;
    const int c = (int)(idx & 63);
    const long long r   = ei[e];
    const long long col = ei[E + e];
    const float a = ea[e];
    const float v = a * Y[col * 128 + 64 + c];   // ea * y2[col][c]
    sgat_atomic_add(&sums[r * 64 + c], v);
    if (c == 0) {
        sgat_atomic_add(&s_ea[r], a);
        sgat_atomic_add(&cnt[r], 1.0f);
    }
}

// ---------------------------------------------------------------------------
// Kernel 4: out[n][c] = (s_ea[n]*y1[n][c] + sums[n][c]) / max(cnt[n],1) + bias[c]
// ---------------------------------------------------------------------------
__global__ void sgat_finalize_kernel(const float* __restrict__ Y,
                                     const float* __restrict__ sums,
                                     const float* __restrict__ s_ea,
                                     const float* __restrict__ cnt,
                                     const float* __restrict__ bias,
                                     float* __restrict__ out, long long N) {
    long long idx = (long long)blockIdx.x * blockDim.x + threadIdx.x;
    if (idx >= N * 64) return;
    const long long n = idx >> 6;
    const int c = (int)(idx & 63);
    const float denom = fmaxf(cnt[n], 1.0f);
    out[idx] = (s_ea[n] * Y[n * 128 + c] + sums[idx]) / denom + bias[c];
}

// ---------------------------------------------------------------------------
// Launch
// ---------------------------------------------------------------------------
extern "C" void kernel_launch(void* const* d_in, const int* in_sizes, int n_in,
                              void* d_out, int out_size, void* d_ws, size_t ws_size,
                              hipStream_t stream) {
    const float*     x    = (const float*)d_in[0];      // [N,64] f32
    const long long* ei   = (const long long*)d_in[1];  // [2,E] int64
    const float*     ea   = (const float*)d_in[2];      // [E] f32
    const float*     w    = (const float*)d_in[3];      // [128,64] f32
    const float*     bias = (const float*)d_in[4];      // [64] f32

    const long long N = in_sizes[0] / 64;
    const long long E = in_sizes[1] / 2;

    float* Y    = (float*)d_ws;        // N*128
    float* sums = Y + N * 128;         // N*64
    float* sea  = sums + N * 64;       // N
    float* cnt  = sea + N;             // N
    float* out  = (float*)d_out;

    // 1) zero sums / s_ea / cnt (contiguous N*66 floats)
    {
        long long nz = N * 66;
        sgat_zero_kernel<<<(unsigned)((nz + 255) / 256), 256, 0, stream>>>(sums, nz);
    }
    // 2) node GEMM: Y = X @ [W_top | W_bot] via f32 WMMA
    {
        dim3 blk(32, 4);
        dim3 grd((unsigned)((N + 15) / 16), 2);
        sgat_gemm_kernel<<<grd, blk, 0, stream>>>(x, w, Y, (int)N);
    }
    // 3) edge scatter (64 threads per edge)
    {
        long long te = E * 64;
        sgat_edge_kernel<<<(unsigned)((te + 255) / 256), 256, 0, stream>>>(
            ei, ea, Y, sums, sea, cnt, E);
    }
    // 4) finalize: mean + bias
    {
        long long tn = N * 64;
        sgat_finalize_kernel<<<(unsigned)((tn + 255) / 256), 256, 0, stream>>>(
            Y, sums, sea, cnt, bias, out, N);
    }
}